// VGAEModel_24000277250672
// MI455X (gfx1250) — compile-verified
//
#include <hip/hip_runtime.h>

typedef __attribute__((ext_vector_type(2))) float v2f;
typedef __attribute__((ext_vector_type(8))) float v8f;
typedef __attribute__((ext_vector_type(4))) int   v4i;

constexpr int kN  = 100000;
constexpr int kE  = 1600000;
constexpr int kIN = 128;
constexpr int kH1 = 128;
constexpr int kH2 = 64;

// ---------------------------------------------------- async global->LDS copy
#if __has_builtin(__builtin_amdgcn_global_load_async_to_lds_b128)
#define HAVE_ASYNC_LDS 1
#else
#define HAVE_ASYNC_LDS 0
#endif

__device__ __forceinline__ void stage_b128(const float* g, float* l) {
#if HAVE_ASYNC_LDS
  __builtin_amdgcn_global_load_async_to_lds_b128(
      (__attribute__((address_space(1))) v4i*)(g),
      (__attribute__((address_space(3))) v4i*)(l), 0, 0);
#else
  *(float4*)l = *(const float4*)g;
#endif
}

__device__ __forceinline__ void wait_async_lds() {
#if HAVE_ASYNC_LDS
#if __has_builtin(__builtin_amdgcn_s_wait_asynccnt)
  __builtin_amdgcn_s_wait_asynccnt(0);
#else
  asm volatile("s_wait_asynccnt 0x0" ::: "memory");
#endif
#endif
}

// ---------------------------------------------------------------- utilities
__global__ void zero_f32(float* __restrict__ p, size_t n) {
  size_t i = (size_t)blockIdx.x * blockDim.x + threadIdx.x;
  if (i < n) p[i] = 0.0f;
}

__global__ void invdeg_k(float* __restrict__ deg, int n) {
  int i = blockIdx.x * blockDim.x + threadIdx.x;
  if (i < n) deg[i] = 1.0f / fmaxf(deg[i], 1.0f);
}

// One wave (32 lanes) per edge; each lane scatters 4 consecutive floats with
// hardware float atomics. Lane 0 optionally accumulates the destination degree.
__global__ __launch_bounds__(256) void scatter_k(const float* __restrict__ x,
                                                 const int* __restrict__ src,
                                                 const int* __restrict__ dst,
                                                 float* __restrict__ agg,
                                                 float* __restrict__ deg,
                                                 int nEdges, int accumDeg) {
  int gtid = blockIdx.x * blockDim.x + threadIdx.x;
  int e    = gtid >> 5;
  int lane = threadIdx.x & 31;
  if (e >= nEdges) return;
  int s = src[e];
  int d = dst[e];
  const float4 v = *(const float4*)(x + (size_t)s * 128 + lane * 4);
  float* base = agg + (size_t)d * 128 + lane * 4;
  unsafeAtomicAdd(base + 0, v.x);
  unsafeAtomicAdd(base + 1, v.y);
  unsafeAtomicAdd(base + 2, v.z);
  unsafeAtomicAdd(base + 3, v.w);
  if (accumDeg && lane == 0) unsafeAtomicAdd(deg + d, 1.0f);
}

// ------------------------------------------------------------ SAGE layer 0
// h[16 x 128] = relu( X @ Ws + (agg * invdeg) @ Wn + b )
// Block: 256 threads = 8 waves; wave w owns output columns [16w, 16w+16).
__global__ __launch_bounds__(256) void sage_layer0(
    const float* __restrict__ X, const float* __restrict__ Agg,
    const float* __restrict__ invdeg, const float* __restrict__ Ws,
    const float* __restrict__ Wn, const float* __restrict__ Bv,
    float* __restrict__ Hout) {
  __shared__ float Xs[16][132];  // padded rows: no LDS bank conflicts
  __shared__ float Gs[16][132];

  const int tid = threadIdx.x;
  const int m0  = blockIdx.x * 16;

  // Async-stage the 16x128 X tile and raw aggregate tile into LDS.
  for (int i = tid; i < 512; i += 256) {  // 512 float4 per matrix
    int r = i >> 5;
    int c = (i & 31) << 2;
    stage_b128(X + (size_t)(m0 + r) * 128 + c, &Xs[r][c]);
    stage_b128(Agg + (size_t)(m0 + r) * 128 + c, &Gs[r][c]);
  }
  wait_async_lds();
  __syncthreads();

  const int wave = tid >> 5;
  const int lane = tid & 31;
  const int row  = lane & 15;   // A: M index / B: N index (within tile)
  const int kh   = lane >> 4;   // K-half selector
  const int n0   = wave << 4;
  const int ncol = n0 + row;

  const float idg = invdeg[m0 + row];  // mean-scale for this lane's A row

  v8f acc;
  const float bias = Bv[ncol];  // C/D: N = lane&15 for every row VGPR
#pragma unroll
  for (int r = 0; r < 8; ++r) acc[r] = bias;

#pragma unroll 4
  for (int kt = 0; kt < 32; ++kt) {
    const int k0 = kt * 4 + kh * 2;
    v2f ax = *(const v2f*)&Xs[row][k0];
    v2f ag = *(const v2f*)&Gs[row][k0];
    ag.x *= idg;
    ag.y *= idg;
    v2f bs, bn;
    bs.x = Ws[(size_t)(k0 + 0) * kH1 + ncol];
    bs.y = Ws[(size_t)(k0 + 1) * kH1 + ncol];
    bn.x = Wn[(size_t)(k0 + 0) * kH1 + ncol];
    bn.y = Wn[(size_t)(k0 + 1) * kH1 + ncol];
    acc = __builtin_amdgcn_wmma_f32_16x16x4_f32(false, ax, false, bs,
                                                (short)0, acc, false, false);
    acc = __builtin_amdgcn_wmma_f32_16x16x4_f32(false, ag, false, bn,
                                                (short)0, acc, false, false);
  }

#pragma unroll
  for (int r = 0; r < 8; ++r) {
    const int m = m0 + r + (kh << 3);
    Hout[(size_t)m * kH1 + ncol] = fmaxf(acc[r], 0.0f);
  }
}

// --------------------------------------------------------- SAGE layers 1+2
// Waves 0-3: mean tile; waves 4-7: log_std tile; then fuse
// out = mean + noise * exp(log_std) for the block's 16x64 slab.
__global__ __launch_bounds__(256) void sage_layer12(
    const float* __restrict__ H, const float* __restrict__ Agg,
    const float* __restrict__ invdeg,
    const float* __restrict__ Ws1, const float* __restrict__ Wn1,
    const float* __restrict__ B1,
    const float* __restrict__ Ws2, const float* __restrict__ Wn2,
    const float* __restrict__ B2,
    const float* __restrict__ noise, float* __restrict__ out) {
  __shared__ float Hs[16][132];
  __shared__ float Gs[16][132];
  __shared__ float Ms[16][64];  // mean tile
  __shared__ float Ls[16][64];  // log_std tile

  const int tid = threadIdx.x;
  const int m0  = blockIdx.x * 16;

  for (int i = tid; i < 512; i += 256) {
    int r = i >> 5;
    int c = (i & 31) << 2;
    stage_b128(H + (size_t)(m0 + r) * 128 + c, &Hs[r][c]);
    stage_b128(Agg + (size_t)(m0 + r) * 128 + c, &Gs[r][c]);
  }
  wait_async_lds();
  __syncthreads();

  const int wave = tid >> 5;
  const int lane = tid & 31;
  const int row  = lane & 15;
  const int kh   = lane >> 4;
  const bool isMean = (wave < 4);
  const int n0   = (wave & 3) << 4;
  const int ncol = n0 + row;

  const float idg = invdeg[m0 + row];

  const float* Wsp = isMean ? Ws1 : Ws2;
  const float* Wnp = isMean ? Wn1 : Wn2;
  const float* Bp  = isMean ? B1 : B2;

  v8f acc;
  const float bias = Bp[ncol];
#pragma unroll
  for (int r = 0; r < 8; ++r) acc[r] = bias;

#pragma unroll 4
  for (int kt = 0; kt < 32; ++kt) {
    const int k0 = kt * 4 + kh * 2;
    v2f ah = *(const v2f*)&Hs[row][k0];
    v2f ag = *(const v2f*)&Gs[row][k0];
    ag.x *= idg;
    ag.y *= idg;
    v2f bs, bn;
    bs.x = Wsp[(size_t)(k0 + 0) * kH2 + ncol];
    bs.y = Wsp[(size_t)(k0 + 1) * kH2 + ncol];
    bn.x = Wnp[(size_t)(k0 + 0) * kH2 + ncol];
    bn.y = Wnp[(size_t)(k0 + 1) * kH2 + ncol];
    acc = __builtin_amdgcn_wmma_f32_16x16x4_f32(false, ah, false, bs,
                                                (short)0, acc, false, false);
    acc = __builtin_amdgcn_wmma_f32_16x16x4_f32(false, ag, false, bn,
                                                (short)0, acc, false, false);
  }

  float* Tile = isMean ? &Ms[0][0] : &Ls[0][0];
#pragma unroll
  for (int r = 0; r < 8; ++r) Tile[(size_t)(r + (kh << 3)) * 64 + ncol] = acc[r];
  __syncthreads();

  // Fuse: out = mean + noise * exp(log_std); 16*64 elems, 4 per thread.
  const int idx = tid * 4;
  const int r   = idx >> 6;
  const int c   = idx & 63;
  const size_t g = (size_t)(m0 + r) * kH2 + c;
  const float4 nz = *(const float4*)(noise + g);
  float4 o;
  o.x = Ms[r][c + 0] + nz.x * expf(Ls[r][c + 0]);
  o.y = Ms[r][c + 1] + nz.y * expf(Ls[r][c + 1]);
  o.z = Ms[r][c + 2] + nz.z * expf(Ls[r][c + 2]);
  o.w = Ms[r][c + 3] + nz.w * expf(Ls[r][c + 3]);
  *(float4*)(out + g) = o;
}

// ------------------------------------------------------------------ launch
extern "C" void kernel_launch(void* const* d_in, const int* in_sizes, int n_in,
                              void* d_out, int out_size, void* d_ws,
                              size_t ws_size, hipStream_t stream) {
  const float* features = (const float*)d_in[0];
  const int*   src      = (const int*)d_in[1];
  const int*   dst      = (const int*)d_in[2];
  const float* noise    = (const float*)d_in[3];
  const float* w_self0  = (const float*)d_in[4];
  const float* w_neigh0 = (const float*)d_in[5];
  const float* b0       = (const float*)d_in[6];
  const float* w_self1  = (const float*)d_in[7];
  const float* w_neigh1 = (const float*)d_in[8];
  const float* b1       = (const float*)d_in[9];
  const float* w_self2  = (const float*)d_in[10];
  const float* w_neigh2 = (const float*)d_in[11];
  const float* b2       = (const float*)d_in[12];
  float* out = (float*)d_out;

  float* agg = (float*)d_ws;                 // N*128 floats
  float* h   = agg + (size_t)kN * kIN;       // N*128 floats
  float* deg = h + (size_t)kN * kH1;         // N floats (later: invdeg)

  const size_t aggN = (size_t)kN * kIN;
  const int scatterBlocks = (int)(((size_t)kE * 32) / 256);

  zero_f32<<<(unsigned)((aggN + 255) / 256), 256, 0, stream>>>(agg, aggN);
  zero_f32<<<(kN + 255) / 256, 256, 0, stream>>>(deg, (size_t)kN);
  scatter_k<<<scatterBlocks, 256, 0, stream>>>(features, src, dst, agg, deg,
                                               kE, 1);
  invdeg_k<<<(kN + 255) / 256, 256, 0, stream>>>(deg, kN);
  sage_layer0<<<kN / 16, 256, 0, stream>>>(features, agg, deg, w_self0,
                                           w_neigh0, b0, h);
  zero_f32<<<(unsigned)((aggN + 255) / 256), 256, 0, stream>>>(agg, aggN);
  scatter_k<<<scatterBlocks, 256, 0, stream>>>(h, src, dst, agg, deg, kE, 0);
  sage_layer12<<<kN / 16, 256, 0, stream>>>(h, agg, deg, w_self1, w_neigh1, b1,
                                            w_self2, w_neigh2, b2, noise, out);
}